// DotInteraction_1082331758806
// MI455X (gfx1250) — compile-verified
//
#include <hip/hip_runtime.h>

typedef __attribute__((ext_vector_type(2))) float v2f;
typedef __attribute__((ext_vector_type(4))) float v4f;
typedef __attribute__((ext_vector_type(8))) float v8f;

#define D_DIM   128
#define S_SPARSE 26
#define N_CAT    27                           // 1 dense + 26 sparse
#define N_TRI   ((N_CAT * (N_CAT + 1)) / 2)   // 378 (tril incl. diagonal)
#define OUT_COLS (D_DIM + N_TRI)              // 506
#define LDS_STRIDE (D_DIM + 4)                // 132 dwords: 132%64==4 -> conflict-free b64 frag loads
#define THREADS 96                            // 3 waves: one per lower-triangle 16x16 tile

__global__ __launch_bounds__(THREADS)
void dot_interaction_kernel(const float* __restrict__ dense,
                            const float* __restrict__ sparse,
                            float* __restrict__ out,
                            int batch) {
  // 32 rows (27 used + 5 pad) x 132 floats = 16.9 KB LDS per workgroup
  __shared__ __align__(16) float lds[32][LDS_STRIDE];

  const int b   = blockIdx.x;
  const int tid = threadIdx.x;

  const float* __restrict__ dense_row = dense + (size_t)b * D_DIM;
  float* __restrict__ out_row = out + (size_t)b * OUT_COLS;

  // ---- Stage cat(b) = [dense_b ; sparse_s,b] into LDS (coalesced 512B rows),
  // ---- and pass the dense row straight through to the output.
  for (int idx = tid; idx < N_CAT * (D_DIM / 4); idx += THREADS) {
    const int row = idx >> 5;            // which of the 27 rows
    const int c4  = (idx & 31) << 2;     // float4 column
    const float* src = (row == 0)
        ? (dense_row + c4)
        : (sparse + ((size_t)(row - 1) * (size_t)batch + (size_t)b) * D_DIM + c4);
    v4f v = *(const v4f*)src;            // global_load_b128
    *(v4f*)&lds[row][c4] = v;            // ds_store_b128
    if (row == 0) *(v4f*)&out_row[c4] = v;
  }
  __syncthreads();

  // ---- One 16x16 WMMA tile per wave: (0,0), (1,0), (1,1) of the 32x32 padded Gram.
  const int wave = tid >> 5;
  const int lane = tid & 31;
  const int half = lane >> 4;   // selects K-pair {2h, 2h+1} within each K=4 step
  const int r    = lane & 15;   // row within tile

  const int mb = (wave == 0) ? 0 : 16;
  const int nb = (wave == 2) ? 16 : 0;

  // Gram symmetry: A (16x4) and B (4x16) fragments use identical per-lane
  // addressing (row base differs). ISA layout: A: V0=K0|K2, V1=K1|K3;
  // B: V0=row K0|K2, V1=row K1|K3 -> lane needs X[base+r][4kk+2h .. +1].
  const float* __restrict__ arow = &lds[mb + r][2 * half];
  const float* __restrict__ brow = &lds[nb + r][2 * half];

  v8f acc = {};
#pragma unroll
  for (int kk = 0; kk < D_DIM / 4; ++kk) {
    v2f af = *(const v2f*)(arow + 4 * kk);   // ds_load_b64
    v2f bf = *(const v2f*)(brow + 4 * kk);   // ds_load_b64
    // D = A x B + C, exact fp32:  v_wmma_f32_16x16x4_f32
    acc = __builtin_amdgcn_wmma_f32_16x16x4_f32(
        false, af, false, bf, (short)0, acc, false, false);
  }

  // ---- Scatter accumulator to packed lower triangle.
  // C/D layout: VGPR v holds M = mb + v + 8*half, N = nb + (lane&15).
  const int n = nb + r;
#pragma unroll
  for (int v = 0; v < 8; ++v) {
    const int m = mb + v + 8 * half;
    if (m < N_CAT && n <= m) {
      out_row[D_DIM + (m * (m + 1)) / 2 + n] = acc[v];
    }
  }
}

extern "C" void kernel_launch(void* const* d_in, const int* in_sizes, int n_in,
                              void* d_out, int out_size, void* d_ws, size_t ws_size,
                              hipStream_t stream) {
  const float* dense  = (const float*)d_in[0];   // (B, 128) fp32
  const float* sparse = (const float*)d_in[1];   // (26, B, 128) fp32
  float* out = (float*)d_out;                    // (B, 506) fp32

  const int batch = in_sizes[0] / D_DIM;         // 16384
  dot_interaction_kernel<<<batch, THREADS, 0, stream>>>(dense, sparse, out, batch);
}